// _DecoderZX_37082747634405
// MI455X (gfx1250) — compile-verified
//
#include <hip/hip_runtime.h>
#include <hip/hip_bf16.h>

#define N_IN      192
#define N_OUT     4096
#define N_NUI     128
#define N_LAT     64
#define BATCH     1024

typedef __attribute__((ext_vector_type(2))) float v2f;
typedef __attribute__((ext_vector_type(8))) float v8f;

// ---------------------------------------------------------------------------
// Kernel A: x1 = z0 (1024x64) @ amat_kernel (64x4096), f32 WMMA 16x16x4.
// One wave32 per 16x16 output tile; K=64 -> 16 WMMA steps.
// D written directly into the mu region of d_out.
// ---------------------------------------------------------------------------
__global__ __launch_bounds__(256) void decoder_x1_wmma(
    const float* __restrict__ z,        // (BATCH, N_IN); z0 = cols [0,64)
    const float* __restrict__ Wk,       // (N_LAT, N_OUT)
    float* __restrict__ mu)             // (BATCH, N_OUT)
{
    const int tiles_n = N_OUT / 16;                       // 256
    const int wave    = (blockIdx.x * blockDim.x + threadIdx.x) >> 5;
    const int lane    = threadIdx.x & 31;
    const int tm      = wave / tiles_n;                   // 0..63
    const int tn      = wave % tiles_n;                   // 0..255
    const int half    = lane >> 4;                        // 0: lanes 0-15, 1: lanes 16-31
    const int l16     = lane & 15;

    v8f c = {};

    const int arow = tm * 16 + l16;                       // M row of A for this lane
    const int bcol = tn * 16 + l16;                       // N col of B for this lane

    #pragma unroll
    for (int k0 = 0; k0 < N_LAT; k0 += 4) {
        // A fragment (16x4 f32): VGPR0 = {K0 | K2}, VGPR1 = {K1 | K3}
        const int ak = k0 + 2 * half;
        v2f a;
        a.x = z[(size_t)arow * N_IN + ak + 0];
        a.y = z[(size_t)arow * N_IN + ak + 1];
        // B fragment (4x16 f32): rows striped across lanes; lanes 16-31 hold rows +2
        const int bk = k0 + 2 * half;
        v2f b;
        b.x = Wk[(size_t)(bk + 0) * N_OUT + bcol];
        b.y = Wk[(size_t)(bk + 1) * N_OUT + bcol];
        // (neg_a, A, neg_b, B, c_mod, C, reuse_a, reuse_b)
        c = __builtin_amdgcn_wmma_f32_16x16x4_f32(false, a, false, b,
                                                  (short)0, c, false, false);
    }

    // D 16x16 f32: lane<16 -> rows M=0..7 (VGPR r = M), lane>=16 -> rows M=8..15
    const int col   = tn * 16 + l16;
    const int rbase = tm * 16 + half * 8;
    #pragma unroll
    for (int r = 0; r < 8; ++r)
        mu[(size_t)(rbase + r) * N_OUT + col] = c[r];
}

// ---------------------------------------------------------------------------
// Kernel B: per batch row (one 256-thread block / 8 waves per row):
//   id = argmax(z[b, 64:192]);  acc = x1 + offsets[id] + z0 @ amat_site[id];
//   softmax over 4096; scale by size_factor[b]; store in place.
// ---------------------------------------------------------------------------
__global__ __launch_bounds__(256) void decoder_row_fused(
    const float* __restrict__ z,         // (BATCH, N_IN)
    const float* __restrict__ size_factor, // (BATCH, 1)
    const float* __restrict__ amat_site, // (N_NUI, N_LAT, N_OUT)
    const float* __restrict__ offsets,   // (N_NUI, N_OUT)
    float* __restrict__ mu)              // (BATCH, N_OUT) in/out
{
    const int b = blockIdx.x;
    const int t = threadIdx.x;

    __shared__ float z0s[N_LAT];
    __shared__ float amax[N_NUI];
    __shared__ int   aidx[N_NUI];
    __shared__ float red[256];

    const float* zrow = z + (size_t)b * N_IN;
    if (t < N_LAT) z0s[t] = zrow[t];
    if (t < N_NUI) { amax[t] = zrow[N_LAT + t]; aidx[t] = t; }
    __syncthreads();

    // argmax reduction (strict > takes the right,
    // ties keep the left, so the first occurrence is not guaranteed)
    #pragma unroll
    for (int s = N_NUI / 2; s >= 1; s >>= 1) {
        if (t < s) {
            if (amax[t + s] > amax[t]) { amax[t] = amax[t + s]; aidx[t] = aidx[t + s]; }
        }
        __syncthreads();
    }
    const int id = aidx[0];

    const float* As  = amat_site + ((size_t)id * N_LAT) * N_OUT;
    const float* off = offsets + (size_t)id * N_OUT;
    float*       murow = mu + (size_t)b * N_OUT;

    // acc[i] covers output column t + 256*i (coalesced across the block)
    float acc[16];
    #pragma unroll
    for (int i = 0; i < 16; ++i)
        acc[i] = murow[t + 256 * i] + off[t + 256 * i];

    for (int l = 0; l < N_LAT; ++l) {
        const float zl = z0s[l];
        const float* Ar = As + (size_t)l * N_OUT;
        #pragma unroll
        for (int i = 0; i < 16; ++i)
            acc[i] = fmaf(zl, Ar[t + 256 * i], acc[i]);
    }

    // row max
    float m = acc[0];
    #pragma unroll
    for (int i = 1; i < 16; ++i) m = fmaxf(m, acc[i]);
    red[t] = m;
    __syncthreads();
    #pragma unroll
    for (int s = 128; s >= 1; s >>= 1) {
        if (t < s) red[t] = fmaxf(red[t], red[t + s]);
        __syncthreads();
    }
    const float rowmax = red[0];
    __syncthreads();

    // exp + row sum
    float sum = 0.0f;
    #pragma unroll
    for (int i = 0; i < 16; ++i) {
        acc[i] = __expf(acc[i] - rowmax);
        sum += acc[i];
    }
    red[t] = sum;
    __syncthreads();
    #pragma unroll
    for (int s = 128; s >= 1; s >>= 1) {
        if (t < s) red[t] += red[t + s];
        __syncthreads();
    }
    const float scale = size_factor[b] / red[0];

    #pragma unroll
    for (int i = 0; i < 16; ++i)
        murow[t + 256 * i] = acc[i] * scale;
}

// ---------------------------------------------------------------------------
// Kernel C: inverse_dispersion = exp(px_r), appended after mu in d_out.
// ---------------------------------------------------------------------------
__global__ __launch_bounds__(256) void decoder_disp(
    const float* __restrict__ px_r, float* __restrict__ out)
{
    const int i = blockIdx.x * blockDim.x + threadIdx.x;
    if (i < N_OUT) out[i] = __expf(px_r[i]);
}

extern "C" void kernel_launch(void* const* d_in, const int* in_sizes, int n_in,
                              void* d_out, int out_size, void* d_ws, size_t ws_size,
                              hipStream_t stream) {
    (void)in_sizes; (void)n_in; (void)out_size; (void)d_ws; (void)ws_size;

    const float* z    = (const float*)d_in[0];  // (1024, 192)
    const float* sf   = (const float*)d_in[1];  // (1024, 1)
    const float* Wk   = (const float*)d_in[2];  // (64, 4096)
    const float* As   = (const float*)d_in[3];  // (128, 64, 4096)
    const float* off  = (const float*)d_in[4];  // (128, 4096)
    const float* pxr  = (const float*)d_in[5];  // (4096,)

    float* mu   = (float*)d_out;                       // (1024, 4096)
    float* disp = mu + (size_t)BATCH * N_OUT;          // (4096,)

    // Kernel A: (1024/16)*(4096/16) = 16384 wave-tiles, 8 waves per 256-thread block.
    const int total_waves = (BATCH / 16) * (N_OUT / 16);
    decoder_x1_wmma<<<total_waves / 8, 256, 0, stream>>>(z, Wk, mu);

    // Kernel B: one block per batch row.
    decoder_row_fused<<<BATCH, 256, 0, stream>>>(z, sf, As, off, mu);

    // Kernel C: dispersion tail.
    decoder_disp<<<N_OUT / 256, 256, 0, stream>>>(pxr, disp);
}